// MeanShiftCluster_76132590289118
// MI455X (gfx1250) — compile-verified
//
#include <hip/hip_runtime.h>
#include <stdint.h>

#define H     224
#define W     224
#define BATCH 4
#define R     9
#define DIA   19
#define TILE  16
#define HALO  (TILE + 2*R)        // 34
#define NLDS  (HALO*HALO)         // 1156
#define NPIX  (BATCH*H*W)         // 200704

// g(d) = exp(-0.5*(d/rsigma)^2) / (rsigma*sqrt(2pi)),  rsigma = 0.5/3
// exp(-18 d^2) = exp2(K2 * d^2),  K2 = -18*log2(e)
#define K2     (-25.968510736f)
#define CNORM  (2.3936536f)       // 1/(rsigma*sqrt(2pi))
#define INVLAB (1.0f/0.51f)       // 1/(rradius+0.01)

// ---- CDNA5 async global->LDS copy (per-lane), tracked by ASYNCcnt ----
__device__ __forceinline__ void async_ld_f32(unsigned lds_byte_off, const float* g) {
  asm volatile("global_load_async_to_lds_b32 %0, %1, off"
               :: "v"(lds_byte_off), "v"((uint64_t)(uintptr_t)g)
               : "memory");
}

__device__ __forceinline__ void wait_async0() {
#if __has_builtin(__builtin_amdgcn_s_wait_asynccnt)
  __builtin_amdgcn_s_wait_asynccnt(0);
#else
  asm volatile("s_wait_asynccnt 0" ::: "memory");
#endif
}

// One mean-shift iteration: 16x16 output tile per block, 34x34 halo tile in LDS.
__global__ __launch_bounds__(256) void ms_iter_kernel(const float* __restrict__ src,
                                                      float* __restrict__ dst,
                                                      int apply_relu) {
  __shared__ float tile[NLDS];
  const int tid = threadIdx.x;
  const int tx  = tid & (TILE - 1);
  const int ty  = tid >> 4;
  const int bx  = blockIdx.x * TILE;
  const int by  = blockIdx.y * TILE;
  const int b   = blockIdx.z;
  const float* img = src + (size_t)b * (H * W);

  // Stage halo tile with edge-replication via async DMA into LDS.
  for (int i = tid; i < NLDS; i += 256) {
    int ly = i / HALO;
    int lx = i - ly * HALO;
    int gy = by + ly - R; gy = gy < 0 ? 0 : (gy > H - 1 ? H - 1 : gy);
    int gx = bx + lx - R; gx = gx < 0 ? 0 : (gx > W - 1 ? W - 1 : gx);
    async_ld_f32((unsigned)(uintptr_t)&tile[i], img + gy * W + gx);
  }
  wait_async0();
  __syncthreads();

  if (apply_relu) {  // fuse relu of the source into iteration 1 (pointwise, tile-consistent)
    for (int i = tid; i < NLDS; i += 256) tile[i] = fmaxf(tile[i], 0.0f);
    __syncthreads();
  }

  const float c = tile[(ty + R) * HALO + (tx + R)];
  float num = 0.0f, den = 0.0f;
  for (int ky = 0; ky < DIA; ++ky) {
    const float* row = &tile[(ty + ky) * HALO + tx];
#pragma unroll
    for (int kx = 0; kx < DIA; ++kx) {
      float n = row[kx];
      float d = c - n;
      float g = CNORM * __builtin_amdgcn_exp2f(K2 * d * d);  // hardware v_exp_f32
      float w = (g < 0.5f) ? 0.1f * g : g;
      num = fmaf(w, n, num);
      den += w;
    }
  }
  dst[(size_t)b * (H * W) + (size_t)(by + ty) * W + (bx + tx)] = num / den;
}

__global__ void init_mm_kernel(int* mm) {
  mm[0] = 0x7fffffff;   // min
  mm[1] = 0x80000000;   // max
}

__global__ __launch_bounds__(256) void label_minmax_kernel(const float* __restrict__ mf,
                                                           int* __restrict__ mm) {
  __shared__ int smin, smax;
  if (threadIdx.x == 0) { smin = 0x7fffffff; smax = (int)0x80000000; }
  __syncthreads();
  int i = blockIdx.x * 256 + threadIdx.x;
  if (i < NPIX) {
    int lab = (int)rintf(mf[i] * INVLAB);
    atomicMin(&smin, lab);
    atomicMax(&smax, lab);
  }
  __syncthreads();
  if (threadIdx.x == 0) {
    atomicMin(&mm[0], smin);
    atomicMax(&mm[1], smax);
  }
}

__global__ __launch_bounds__(256) void label_final_kernel(const float* __restrict__ mf,
                                                          const int* __restrict__ mm,
                                                          int* __restrict__ out) {
  int i = blockIdx.x * 256 + threadIdx.x;
  if (i < NPIX) {
    int lab = (int)rintf(mf[i] * INVLAB);
    float mn = (float)mm[0];
    float mx = (float)mm[1];
    out[i] = (int)rintf(((float)lab - mn) / (mx - mn) * 254.0f);
  }
}

extern "C" void kernel_launch(void* const* d_in, const int* in_sizes, int n_in,
                              void* d_out, int out_size, void* d_ws, size_t ws_size,
                              hipStream_t stream) {
  const float* x = (const float*)d_in[0];
  float* buf0 = (float*)d_ws;
  float* buf1 = buf0 + NPIX;
  int*   mm   = (int*)(buf1 + NPIX);
  int*   out  = (int*)d_out;

  dim3 grid(W / TILE, H / TILE, BATCH);   // 14 x 14 x 4 = 784 workgroups
  dim3 blk(256);                           // 8 wave32 per block

  ms_iter_kernel<<<grid, blk, 0, stream>>>(x,    buf0, 1);  // fused relu
  ms_iter_kernel<<<grid, blk, 0, stream>>>(buf0, buf1, 0);
  ms_iter_kernel<<<grid, blk, 0, stream>>>(buf1, buf0, 0);
  ms_iter_kernel<<<grid, blk, 0, stream>>>(buf0, buf1, 0);

  init_mm_kernel<<<1, 1, 0, stream>>>(mm);
  const int nblk = (NPIX + 255) / 256;     // 784
  label_minmax_kernel<<<nblk, 256, 0, stream>>>(buf1, mm);
  label_final_kernel<<<nblk, 256, 0, stream>>>(buf1, mm, out);
}